// LinkPredictor_42863773614348
// MI455X (gfx1250) — compile-verified
//
#include <hip/hip_runtime.h>

typedef float v2f __attribute__((ext_vector_type(2)));
typedef float v8f __attribute__((ext_vector_type(8)));

#define Bsz 4
#define Nn  256
#define Tt  16
#define Ff  128
#define Hh  256

// ---------------------------------------------------------------------------
// Stage 1: h[m][f] = mean over T of nodefeat[m][t][f], m in [0, B*N)
// ---------------------------------------------------------------------------
__global__ void lp_mean_kernel(const float* __restrict__ nf, float* __restrict__ h) {
    int idx = blockIdx.x * blockDim.x + threadIdx.x;   // 0 .. B*N*F-1
    int m = idx >> 7;          // / F
    int f = idx & (Ff - 1);    // % F
    const float* p = nf + m * (Tt * Ff) + f;
    float s = 0.0f;
#pragma unroll
    for (int t = 0; t < Tt; ++t) s += p[t * Ff];
    h[idx] = s * (1.0f / (float)Tt);
}

// ---------------------------------------------------------------------------
// Stage 2: fp32 WMMA GEMM, one wave per 16x16 output tile.
//   z==0:  a[m][ho]  = sum_f h[m][f] * W1[ho][f]        + b1[ho]
//   z==1:  ct[b][ho][j] = sum_f h[b*N+j][f] * W1[ho][F+f]   (stored transposed)
// A-frag layout (16x4 f32): lane<16 -> row=lane, K=k..k+1 ; lane>=16 -> K=k+2..k+3
// B-frag layout (4x16 f32): lane<16 -> col=lane, K=k..k+1 ; lane>=16 -> K=k+2..k+3
// C/D layout: VGPR r -> M = r + 8*(lane/16), N = lane%16
// ---------------------------------------------------------------------------
__global__ void lp_gemm_wmma(const float* __restrict__ h,    // [B*N, F]
                             const float* __restrict__ W1,   // [H, 2F] row-major
                             const float* __restrict__ b1,   // [H]
                             float* __restrict__ a_out,      // [B*N, H]
                             float* __restrict__ ct_out) {   // [B, H, N]
    const int row0   = blockIdx.x * 16;       // m tile
    const int col0   = blockIdx.y * 16;       // ho tile
    const int colOff = blockIdx.z * Ff;       // 0 -> W1[:, :F], 1 -> W1[:, F:]
    const int lane   = threadIdx.x;           // 0..31, wave32
    const int half   = lane >> 4;             // 0|1 selects K sub-pair
    const int lmod   = lane & 15;             // row (A) / col (B) within tile

    const float* aptr = h  + (row0 + lmod) * Ff          + half * 2;
    const float* bptr = W1 + (col0 + lmod) * (2 * Ff) + colOff + half * 2;

    v8f acc = {};
#pragma unroll
    for (int k = 0; k < Ff; k += 4) {
        v2f af = *(const v2f*)(aptr + k);
        v2f bf = *(const v2f*)(bptr + k);
        acc = __builtin_amdgcn_wmma_f32_16x16x4_f32(
            /*neg_a=*/false, af, /*neg_b=*/false, bf,
            /*c_mod=*/(short)0, acc, /*reuse_a=*/false, /*reuse_b=*/false);
    }

    if (blockIdx.z == 0) {
        const float bias = b1[col0 + lmod];
#pragma unroll
        for (int r = 0; r < 8; ++r) {
            int m = row0 + half * 8 + r;
            a_out[m * Hh + col0 + lmod] = acc[r] + bias;
        }
    } else {
#pragma unroll
        for (int r = 0; r < 8; ++r) {
            int m  = row0 + half * 8 + r;   // global node index b*N + j
            int bb = m >> 8;
            int j  = m & (Nn - 1);
            ct_out[bb * (Hh * Nn) + (col0 + lmod) * Nn + j] = acc[r];
        }
    }
}

// ---------------------------------------------------------------------------
// Stage 3 (fused): logits[b][i][j] = b2 + sum_h w2[h]*relu(a[b,i,h]+ct[b,h,j])
// Block = 256 threads (8 waves), one block per (b, i). a-row & w2 in LDS.
// ct reads are lane-coalesced (lane == j).
// ---------------------------------------------------------------------------
__global__ void lp_pair_score(const float* __restrict__ a,   // [B*N, H] (b1 folded in)
                              const float* __restrict__ ct,  // [B, H, N]
                              const float* __restrict__ W2,  // [H]
                              const float* __restrict__ b2,  // [1]
                              float* __restrict__ out) {     // [B, N, N]
    __shared__ float as[Hh];
    __shared__ float w2s[Hh];
    const int j = threadIdx.x;
    const int i = blockIdx.x;
    const int b = blockIdx.y;

    as[j]  = a[(b * Nn + i) * Hh + j];
    w2s[j] = W2[j];
    __syncthreads();

    const float* cp = ct + b * (Hh * Nn) + j;   // stride Nn per h, coalesced per lane
    float acc = 0.0f;
#pragma unroll 8
    for (int hh = 0; hh < Hh; ++hh) {
        float v = as[hh] + cp[hh * Nn];
        acc = fmaf(w2s[hh], fmaxf(v, 0.0f), acc);
    }
    out[(b * Nn + i) * Nn + j] = acc + b2[0];
}

// ---------------------------------------------------------------------------
extern "C" void kernel_launch(void* const* d_in, const int* in_sizes, int n_in,
                              void* d_out, int out_size, void* d_ws, size_t ws_size,
                              hipStream_t stream) {
    const float* nodefeat = (const float*)d_in[0];   // [B,N,T,F]
    const float* W1       = (const float*)d_in[1];   // [H, 2F]
    const float* b1       = (const float*)d_in[2];   // [H]
    const float* W2       = (const float*)d_in[3];   // [1, H]
    const float* b2       = (const float*)d_in[4];   // [1]
    float* out = (float*)d_out;

    float* ws = (float*)d_ws;
    float* h  = ws;                               // B*N*F   = 131072 floats
    float* a  = ws + (Bsz * Nn * Ff);             // B*N*H   = 262144 floats
    float* ct = a  + (Bsz * Nn * Hh);             // B*H*N   = 262144 floats
    // total workspace: 2.5 MB

    // Stage 1: temporal mean
    lp_mean_kernel<<<dim3((Bsz * Nn * Ff) / 256), 256, 0, stream>>>(nodefeat, h);

    // Stage 2: a and c^T via fp32 WMMA (one wave per 16x16 tile)
    lp_gemm_wmma<<<dim3((Bsz * Nn) / 16, Hh / 16, 2), 32, 0, stream>>>(h, W1, b1, a, ct);

    // Stage 3: fused relu + dot(w2) pair scoring
    lp_pair_score<<<dim3(Nn, Bsz), 256, 0, stream>>>(a, ct, W2, b2, out);
}